// CategoryConsistencyLoss_6777458393605
// MI455X (gfx1250) — compile-verified
//
#include <hip/hip_runtime.h>

typedef __attribute__((ext_vector_type(2))) float v2f;
typedef __attribute__((ext_vector_type(8))) float v8f;

#define N_ROWS    16384
#define DIM       2048
#define BLK       256   // 8 wave32s

// One block per row: dist[row] = clip(||x_row - W[label_row]||^2, 1e-12, 1e12)
__global__ __launch_bounds__(BLK) void ccl_row_dist_kernel(
    const float* __restrict__ x,
    const int*   __restrict__ labels,
    const float* __restrict__ w,
    float*       __restrict__ row_dist)
{
    const int row = blockIdx.x;
    const int tid = threadIdx.x;
    const int lbl = labels[row];

    const float4* __restrict__ xr =
        reinterpret_cast<const float4*>(x + (size_t)row * DIM);
    const float4* __restrict__ wr =
        reinterpret_cast<const float4*>(w + (size_t)lbl * DIM);

    // 2048 floats = 512 float4; 256 threads -> 2 float4 per thread (b128 loads)
    float acc = 0.0f;
#pragma unroll
    for (int i = 0; i < 2; ++i) {
        const int idx = tid + i * BLK;
        const float4 xv = xr[idx];
        const float4 wv = wr[idx];
        const float d0 = xv.x - wv.x;
        const float d1 = xv.y - wv.y;
        const float d2 = xv.z - wv.z;
        const float d3 = xv.w - wv.w;
        acc = fmaf(d0, d0, acc);
        acc = fmaf(d1, d1, acc);
        acc = fmaf(d2, d2, acc);
        acc = fmaf(d3, d3, acc);
    }

    // ---- wave32 cross-lane reduction via V_WMMA_F32_16X16X4_F32 ----
    // A (16x4 fp32): lanes 0-15 supply K=0 (VGPR0) / K=1 (VGPR1),
    //                lanes 16-31 supply K=2 / K=3.
    // Put partial in K-even slot, 0 in K-odd slot; B = all ones; C = 0.
    // => D[m][n] = p[m] + p[m+16] for every n.
    v2f a; a[0] = acc;  a[1] = 0.0f;
    v2f b; b[0] = 1.0f; b[1] = 1.0f;
    v8f c = {};
    c = __builtin_amdgcn_wmma_f32_16x16x4_f32(
            /*neg_a=*/false, a, /*neg_b=*/false, b,
            /*c_mod=*/(short)0, c, /*reuse_a=*/false, /*reuse_b=*/false);

    // Each lane sums its 8 D VGPRs:
    //   lanes 0-15  hold D[0..7][n]  -> sum_{m=0..7}(p[m]+p[m+16])
    //   lanes 16-31 hold D[8..15][n] -> sum_{m=8..15}(p[m]+p[m+16])
    // wave total = (any low-half lane) + (any high-half lane)
    float s = c[0] + c[1] + c[2] + c[3] + c[4] + c[5] + c[6] + c[7];

    __shared__ float lds[(BLK / 32) * 2];
    const int wave = tid >> 5;
    const int lane = tid & 31;
    if (lane == 0)  lds[wave * 2 + 0] = s;
    if (lane == 16) lds[wave * 2 + 1] = s;
    __syncthreads();

    if (tid == 0) {
        float dist = 0.0f;
#pragma unroll
        for (int i = 0; i < (BLK / 32) * 2; ++i) dist += lds[i];
        dist = fminf(fmaxf(dist, 1e-12f), 1e12f);
        row_dist[row] = dist;
    }
}

// Deterministic final reduction: loss = sum(row_dist) / N_ROWS
__global__ __launch_bounds__(BLK) void ccl_reduce_kernel(
    const float* __restrict__ row_dist,
    float*       __restrict__ out)
{
    __shared__ double lds[BLK];
    double acc = 0.0;
    for (int i = threadIdx.x; i < N_ROWS; i += BLK)
        acc += (double)row_dist[i];
    lds[threadIdx.x] = acc;
    __syncthreads();
#pragma unroll
    for (int s = BLK / 2; s > 0; s >>= 1) {
        if ((int)threadIdx.x < s) lds[threadIdx.x] += lds[threadIdx.x + s];
        __syncthreads();
    }
    if (threadIdx.x == 0)
        out[0] = (float)(lds[0] / (double)N_ROWS);
}

extern "C" void kernel_launch(void* const* d_in, const int* in_sizes, int n_in,
                              void* d_out, int out_size, void* d_ws, size_t ws_size,
                              hipStream_t stream)
{
    (void)in_sizes; (void)n_in; (void)out_size; (void)ws_size;

    const float* x      = (const float*)d_in[0];   // [16384, 2048] f32
    const int*   labels = (const int*)  d_in[1];   // [16384] i32
    const float* w      = (const float*)d_in[2];   // [1000, 2048] f32

    float* row_dist = (float*)d_ws;                // 16384 floats = 64 KB scratch

    ccl_row_dist_kernel<<<N_ROWS, BLK, 0, stream>>>(x, labels, w, row_dist);
    ccl_reduce_kernel<<<1, BLK, 0, stream>>>(row_dist, (float*)d_out);
}